// AIGStateEncoder_56530359550737
// MI455X (gfx1250) — compile-verified
//
#include <hip/hip_runtime.h>
#include <cstddef>

// ---------------------------------------------------------------------------
// AIG state-encoder GNN forward for MI455X (gfx1250), full fp32 via
// V_WMMA_F32_16X16X4_F32 (exact fp32 matrix math, 2048 FLOP/instr).
// Wave32; 16x16 tiles; A-frag a[v]=A[l&15][2*(l>>4)+v], B-frag
// b[v]=W[n0+(l&15)][2*(l>>4)+v+k] (W row-major [N,K] == B^T), C/D
// c[r]=D[r+8*(l>>4)][l&15].
// ---------------------------------------------------------------------------

typedef __attribute__((ext_vector_type(2))) float v2f;
typedef __attribute__((ext_vector_type(8))) float v8f;

static constexpr int   BGRAPH = 512;
static constexpr int   NPG    = 400;
static constexpr int   HC     = 128;
static constexpr int   NTOT   = BGRAPH * NPG;   // 204800 nodes
static constexpr int   ETOT   = 2 * NTOT;       // 409600 edges
static constexpr float SCALE  = 0.08838834764831845f; // 1/sqrt(128)

__device__ __forceinline__ v8f wmma4(v2f a, v2f b, v8f c) {
  return __builtin_amdgcn_wmma_f32_16x16x4_f32(false, a, false, b, (short)0, c,
                                               false, false);
}

// ---------------- node encoder: x4[i] = {emb[type], inv} ----------------
__global__ void encode_kernel(const int* __restrict__ nt,
                              const float* __restrict__ ninv,
                              const float* __restrict__ emb,
                              float* __restrict__ x4, int n) {
  int i = blockIdx.x * blockDim.x + threadIdx.x;
  if (i >= n) return;
  int t = nt[i];
  x4[i * 4 + 0] = emb[t * 3 + 0];
  x4[i * 4 + 1] = emb[t * 3 + 1];
  x4[i * 4 + 2] = emb[t * 3 + 2];
  x4[i * 4 + 3] = ninv[i];
}

// ---- softmax aggregation (shift-free: scores bounded): num/den atomics ----
template <int C>
__global__ void edge_softmax_msg(const float* __restrict__ x,
                                 const int* __restrict__ ei,
                                 const float* __restrict__ tp,
                                 float* __restrict__ num,
                                 float* __restrict__ den, int ne) {
  long long idx = (long long)blockIdx.x * blockDim.x + threadIdx.x;
  if (idx >= (long long)ne * C) return;
  int e = (int)(idx / C), c = (int)(idx % C);
  int s = ei[e], d = ei[ne + e];
  float m = x[(size_t)s * C + c];
  float w = __expf(tp[0] * m);
  atomicAdd(den + (size_t)d * C + c, w);
  atomicAdd(num + (size_t)d * C + c, w * m);
}

__global__ void finalize_agg(float* __restrict__ num,
                             const float* __restrict__ den, long long total) {
  long long i = (long long)blockIdx.x * blockDim.x + threadIdx.x;
  if (i >= total) return;
  float d = den[i];
  num[i] = d > 0.f ? num[i] / d : 0.f;
}

// ---------------- conv1 linear (K=4, plain VALU) ----------------
__global__ void conv1_linear(const float* __restrict__ agg,
                             const float* __restrict__ x4,
                             const float* __restrict__ wl,
                             const float* __restrict__ wr,
                             const float* __restrict__ bvec,
                             float* __restrict__ out, int n) {
  int idx = blockIdx.x * blockDim.x + threadIdx.x;
  if (idx >= n * HC) return;
  int i = idx >> 7, h = idx & 127;
  const float* a = agg + (size_t)i * 4;
  const float* x = x4 + (size_t)i * 4;
  const float* wlr = wl + h * 4;
  const float* wrr = wr + h * 4;
  float v = bvec[h];
#pragma unroll
  for (int c = 0; c < 4; ++c) v += a[c] * wlr[c] + x[c] * wrr[c];
  out[idx] = fmaxf(v, 0.f);
}

// ---------------- generic fp32 WMMA GEMM: D = A @ W^T (+epilogue) ----------
// grid: (M/16, ceil(Ntiles/8), [ksplit]); block 256 (8 waves, one n-tile each)
template <bool DUAL, bool RELU, bool RES, bool SPLIT>
__global__ __launch_bounds__(256) void gemm_f32(
    const float* __restrict__ A, int lda, const float* __restrict__ W, int ldw,
    const float* __restrict__ A2, const float* __restrict__ W2,
    const float* __restrict__ bias, const float* __restrict__ bias2,
    const float* __restrict__ res, int ldres, float* __restrict__ D, int ldd,
    int M, int N, int K, int kChunk) {
  (void)M;
  int w = threadIdx.x >> 5;
  int nt = blockIdx.y * 8 + w;
  if (nt * 16 >= N) return;
  int lane = threadIdx.x & 31;
  int col = lane & 15;   // A row / W row / D col within tile
  int grp = lane >> 4;   // k sub-offset selector
  int m0 = blockIdx.x * 16, n0 = nt * 16;
  int k0 = 0, k1 = K;
  if (SPLIT) {
    k0 = blockIdx.z * kChunk;
    k1 = k0 + kChunk;
    if (k1 > K) k1 = K;
  }
  const float* Ap = A + (size_t)(m0 + col) * lda + 2 * grp;
  const float* Wp = W + (size_t)(n0 + col) * ldw + 2 * grp;
  v8f c = {};
#pragma unroll 8
  for (int k = k0; k < k1; k += 4) {
    v2f a = *(const v2f*)(Ap + k);
    v2f b = *(const v2f*)(Wp + k);
    c = wmma4(a, b, c);
  }
  if (DUAL) {
    const float* A2p = A2 + (size_t)(m0 + col) * lda + 2 * grp;
    const float* W2p = W2 + (size_t)(n0 + col) * ldw + 2 * grp;
#pragma unroll 8
    for (int k = k0; k < k1; k += 4) {
      v2f a = *(const v2f*)(A2p + k);
      v2f b = *(const v2f*)(W2p + k);
      c = wmma4(a, b, c);
    }
  }
  int ocol = n0 + col;
  float bval = 0.f;
  bool addb = !SPLIT || (blockIdx.z == 0);
  if (addb) {
    if (bias) bval += bias[ocol];
    if (bias2) bval += bias2[ocol];
  }
#pragma unroll
  for (int r = 0; r < 8; ++r) {
    int orow = m0 + r + 8 * grp;
    float v = c[r] + bval;
    if (RELU) v = fmaxf(v, 0.f);
    if (RES) v += res[(size_t)orow * ldres + ocol];
    if (SPLIT)
      atomicAdd(&D[(size_t)orow * ldd + ocol], v);
    else
      D[(size_t)orow * ldd + ocol] = v;
  }
}

// ------- SAB encoder attention: per (graph, 16-row q-tile), one wave -------
// O[16,128] = softmax(Q_tile @ K^T / sqrt128) @ V   (400 keys, no mask needed)
__global__ __launch_bounds__(32) void attn_kernel(const float* __restrict__ Q,
                                                  const float* __restrict__ Km,
                                                  const float* __restrict__ V,
                                                  float* __restrict__ O) {
  int qt = blockIdx.x;  // 0..24
  int b = blockIdx.y;   // 0..511
  int lane = threadIdx.x;
  int col = lane & 15, grp = lane >> 4;
  __shared__ __align__(16) float qs[16 * 128];
  __shared__ float et[16][17];
  __shared__ float den[16];
  const float* Qb = Q + ((size_t)b * NPG + qt * 16) * HC;
  const float* Kb = Km + (size_t)b * NPG * HC;
  const float* Vb = V + (size_t)b * NPG * HC;
  for (int i = lane * 4; i < 16 * 128; i += 32 * 4)
    *(float4*)(qs + i) = *(const float4*)(Qb + i);
  if (lane < 16) den[lane] = 0.f;
  __syncthreads();
  v8f acc[8];
#pragma unroll
  for (int i = 0; i < 8; ++i) acc[i] = {};
  for (int kt = 0; kt < 25; ++kt) {
    v8f s = {};
    const float* Kp = Kb + ((size_t)kt * 16 + col) * HC + 2 * grp;
    const float* qrow = qs + col * HC + 2 * grp;
#pragma unroll 8
    for (int k = 0; k < HC; k += 4) {
      v2f a;
      a.x = qrow[k];
      a.y = qrow[k + 1];
      v2f bb = *(const v2f*)(Kp + k);
      s = wmma4(a, bb, s);
    }
    __syncthreads();  // previous-iteration readers of et are done
#pragma unroll
    for (int r = 0; r < 8; ++r) et[r + 8 * grp][col] = __expf(s[r] * SCALE);
    __syncthreads();
    if (lane < 16) {
      float ss = 0.f;
#pragma unroll
      for (int j = 0; j < 16; ++j) ss += et[lane][j];
      den[lane] += ss;
    }
    const float* Vbase = Vb + ((size_t)kt * 16 + 2 * grp) * HC + col;
#pragma unroll
    for (int nt = 0; nt < 8; ++nt) {
      v8f cc = acc[nt];
      const float* Vp = Vbase + nt * 16;
#pragma unroll
      for (int k = 0; k < 16; k += 4) {
        v2f a;
        a.x = et[col][k + 2 * grp];
        a.y = et[col][k + 2 * grp + 1];
        v2f bb;
        bb.x = Vp[(size_t)k * HC];
        bb.y = Vp[(size_t)(k + 1) * HC];
        cc = wmma4(a, bb, cc);
      }
      acc[nt] = cc;
    }
  }
  __syncthreads();
  float* Ob = O + ((size_t)b * NPG + qt * 16) * HC;
#pragma unroll
  for (int r = 0; r < 8; ++r) {
    float dinv = 1.f / den[r + 8 * grp];
    int orow = r + 8 * grp;
#pragma unroll
    for (int nt = 0; nt < 8; ++nt)
      Ob[(size_t)orow * HC + nt * 16 + col] = acc[nt][r] * dinv;
  }
}

// ---------------- GRU over 500 steps; 16 graphs per block -------------------
// gi_all already holds xd@gwih^T + gbih for n<400; t>=400 uses gbih alone.
__global__ __launch_bounds__(256) void gru_kernel(
    const float* __restrict__ gi_all, const float* __restrict__ gwhh,
    const float* __restrict__ gbhh, const float* __restrict__ gbih,
    float* __restrict__ pooled) {
  int b0g = blockIdx.x * 16;
  __shared__ float hs[16 * 128];
  __shared__ float gh[16 * 384];
  int tid = threadIdx.x;
  for (int i = tid; i < 16 * 128; i += 256) hs[i] = 0.f;
  __syncthreads();
  int lane = tid & 31, w = tid >> 5;
  int col = lane & 15, grp = lane >> 4;
  for (int t = 0; t < 500; ++t) {
    // gh = h @ gwhh^T + gbhh  (M=16, N=384, K=128) via WMMA
    for (int nt = w; nt < 24; nt += 8) {
      int n0 = nt * 16;
      v8f c = {};
      const float* hrow = hs + col * 128 + 2 * grp;
      const float* Wp = gwhh + (size_t)(n0 + col) * 128 + 2 * grp;
#pragma unroll 8
      for (int k = 0; k < 128; k += 4) {
        v2f a;
        a.x = hrow[k];
        a.y = hrow[k + 1];
        v2f bb = *(const v2f*)(Wp + k);
        c = wmma4(a, bb, c);
      }
      int ocol = n0 + col;
#pragma unroll
      for (int r = 0; r < 8; ++r)
        gh[(r + 8 * grp) * 384 + ocol] = c[r] + gbhh[ocol];
    }
    __syncthreads();
    for (int i = tid; i < 16 * 128; i += 256) {
      int m = i >> 7, h = i & 127;
      float ir, iz, inn;
      if (t < NPG) {
        const float* gip = gi_all + ((size_t)(b0g + m) * NPG + t) * 384;
        ir = gip[h];
        iz = gip[128 + h];
        inn = gip[256 + h];
      } else {
        ir = gbih[h];
        iz = gbih[128 + h];
        inn = gbih[256 + h];
      }
      float hr = gh[m * 384 + h];
      float hz = gh[m * 384 + 128 + h];
      float hn = gh[m * 384 + 256 + h];
      float r_ = 1.f / (1.f + __expf(-(ir + hr)));
      float z_ = 1.f / (1.f + __expf(-(iz + hz)));
      float n_ = tanhf(inn + r_ * hn);
      float hp = hs[i];
      hs[i] = (1.f - z_) * n_ + z_ * hp;
    }
    __syncthreads();
  }
  for (int i = tid; i < 16 * 128; i += 256) {
    int m = i >> 7, h = i & 127;
    pooled[(size_t)(b0g + m) * 384 + 128 + h] = hs[i];
  }
}

// --------- PMA: scores vs shared seed-query, softmax over 400, context ------
__global__ __launch_bounds__(128) void pma_ctx_kernel(
    const float* __restrict__ kp, const float* __restrict__ vp,
    const float* __restrict__ seed, const float* __restrict__ w0,
    const float* __restrict__ b0, float* __restrict__ ctx) {
  int b = blockIdx.x;
  int tid = threadIdx.x;  // 128
  __shared__ float qp[128];
  __shared__ float es[NPG];
  __shared__ float ssum;
  {
    float acc = b0[tid];
    const float* wr = w0 + (size_t)tid * 128;
    for (int k = 0; k < 128; ++k) acc += seed[k] * wr[k];
    qp[tid] = acc;
  }
  if (tid == 0) ssum = 0.f;
  __syncthreads();
  float part = 0.f;
  for (int n = tid; n < NPG; n += 128) {
    const float* kr = kp + ((size_t)b * NPG + n) * HC;
    float s = 0.f;
    for (int k = 0; k < 128; ++k) s += qp[k] * kr[k];
    float e = __expf(s * SCALE);
    es[n] = e;
    part += e;
  }
  atomicAdd(&ssum, part);
  __syncthreads();
  float inv = 1.f / ssum;
  float a = 0.f;
  for (int n = 0; n < NPG; ++n)
    a += es[n] * vp[((size_t)b * NPG + n) * HC + tid];
  ctx[(size_t)b * HC + tid] = a * inv;
}

// ---------------------------------------------------------------------------
extern "C" void kernel_launch(void* const* d_in, const int* in_sizes, int n_in,
                              void* d_out, int out_size, void* d_ws,
                              size_t ws_size, hipStream_t stream) {
  (void)in_sizes; (void)n_in; (void)out_size; (void)ws_size;
  const int* node_type = (const int*)d_in[0];
  const float* node_inv = (const float*)d_in[1];
  const int* edge_index = (const int*)d_in[2];
  const float* emb = (const float*)d_in[3];
  const float* tptr = (const float*)d_in[4];
  const float* c1wl = (const float*)d_in[5];
  const float* c1wr = (const float*)d_in[6];
  const float* c1b = (const float*)d_in[7];
  const float* c2wl = (const float*)d_in[8];
  const float* c2wr = (const float*)d_in[9];
  const float* c2b = (const float*)d_in[10];
  const float* mlp_w = (const float*)d_in[11];
  const float* mlp_b = (const float*)d_in[12];
  const float* gwih = (const float*)d_in[13];
  const float* gwhh = (const float*)d_in[14];
  const float* gbih = (const float*)d_in[15];
  const float* gbhh = (const float*)d_in[16];
  const float* enc_w = (const float*)d_in[17];
  const float* enc_b = (const float*)d_in[18];
  const float* enc_lw = (const float*)d_in[19];
  const float* enc_lb = (const float*)d_in[20];
  const float* pma_lw = (const float*)d_in[21];
  const float* pma_lb = (const float*)d_in[22];
  const float* pma_seed = (const float*)d_in[23];
  const float* pma_w = (const float*)d_in[24];
  const float* pma_b = (const float*)d_in[25];
  const float* pma_mlw = (const float*)d_in[26];
  const float* pma_mlb = (const float*)d_in[27];
  const float* dec_w = (const float*)d_in[28];
  const float* dec_b = (const float*)d_in[29];
  const float* dec_lw = (const float*)d_in[30];
  const float* dec_lb = (const float*)d_in[31];
  const float* out_w = (const float*)d_in[32];
  const float* out_b = (const float*)d_in[33];

  // Workspace: 6 reused N*128 planes (104.9 MB each) + small region (~12 MB)
  float* ws = (float*)d_ws;
  const size_t PLANE = (size_t)NTOT * HC;
  float* P0 = ws;                 // h1 -> GI(P0..P2) -> Q -> r1(xe)
  float* P1 = ws + 1 * PLANE;     // num2/agg2 -> K -> y
  float* P2 = ws + 2 * PLANE;     // den2 -> V -> kp
  float* P3 = ws + 3 * PLANE;     // XH (conv2 out, residual, MLP-pool A)
  float* P4 = ws + 4 * PLANE;     // attn -> vp
  float* P5 = ws + 5 * PLANE;     // u
  float* SM = ws + 6 * PLANE;
  float* x4 = SM;                               // N*4
  float* num1 = SM + (size_t)NTOT * 4;          // N*4
  float* den1 = SM + (size_t)NTOT * 8;          // N*4
  float* pooled = SM + (size_t)NTOT * 12;       // 512*384
  float* ctx = pooled + 512 * 384;              // 512*128
  float* xp_pre = ctx + 512 * 128;
  float* xp = xp_pre + 512 * 128;
  float* vdec = xp + 512 * 128;
  float* xs_pre = vdec + 512 * 128;

  auto cdiv = [](long long a, long long b) { return (int)((a + b - 1) / b); };

  // zero accumulation buffers (every call: ws is not re-poisoned)
  hipMemsetAsync(num1, 0, (size_t)NTOT * 8 * sizeof(float), stream);
  hipMemsetAsync(P1, 0, 2 * PLANE * sizeof(float), stream);
  hipMemsetAsync(pooled, 0, 512 * 384 * sizeof(float), stream);

  // --- node encoder + conv1 (softmax aggregation, K=4 linear) ---
  encode_kernel<<<cdiv(NTOT, 256), 256, 0, stream>>>(node_type, node_inv, emb,
                                                     x4, NTOT);
  edge_softmax_msg<4><<<cdiv((long long)ETOT * 4, 256), 256, 0, stream>>>(
      x4, edge_index, tptr, num1, den1, ETOT);
  finalize_agg<<<cdiv((long long)NTOT * 4, 256), 256, 0, stream>>>(
      num1, den1, (long long)NTOT * 4);
  conv1_linear<<<cdiv((long long)NTOT * HC, 256), 256, 0, stream>>>(
      num1, x4, c1wl, c1wr, c1b, P0, NTOT);

  // --- conv2: aggregation + dual-GEMM relu(agg@wl^T + h1@wr^T + b) -> XH ---
  edge_softmax_msg<128><<<cdiv((long long)ETOT * 128, 256), 256, 0, stream>>>(
      P0, edge_index, tptr, P1, P2, ETOT);
  finalize_agg<<<cdiv((long long)PLANE, 256), 256, 0, stream>>>(
      P1, P2, (long long)PLANE);
  gemm_f32<true, true, false, false><<<dim3(NTOT / 16, 1), 256, 0, stream>>>(
      P1, 128, c2wl, 128, P0, c2wr, c2b, nullptr, nullptr, 0, P3, 128, NTOT,
      128, 128, 0);

  // --- GRU input GEMM: GI = XH@gwih^T + gbih  (N x 384, occupies P0..P2) ---
  gemm_f32<false, false, false, false><<<dim3(NTOT / 16, 3), 256, 0, stream>>>(
      P3, 128, gwih, 128, nullptr, nullptr, gbih, nullptr, nullptr, 0, P0, 384,
      NTOT, 384, 128, 0);
  gru_kernel<<<32, 256, 0, stream>>>(P0, gwhh, gbhh, gbih, pooled);

  // --- MLP pool: p1 = flat(xd)@mlp_w^T + mlp_b  (split-K, atomic) ---
  gemm_f32<false, false, false, true><<<dim3(32, 1, 16), 256, 0, stream>>>(
      P3, NPG * HC, mlp_w, 500 * HC, nullptr, nullptr, mlp_b, nullptr, nullptr,
      0, pooled, 384, 512, 128, NPG * HC, 3200);

  // --- SAB encoder: Q/K/V projections, attention, out-proj + residual ---
  gemm_f32<false, false, false, false><<<dim3(NTOT / 16, 1), 256, 0, stream>>>(
      P3, 128, enc_w + 0 * 16384, 128, nullptr, nullptr, enc_b + 0, nullptr,
      nullptr, 0, P0, 128, NTOT, 128, 128, 0);
  gemm_f32<false, false, false, false><<<dim3(NTOT / 16, 1), 256, 0, stream>>>(
      P3, 128, enc_w + 1 * 16384, 128, nullptr, nullptr, enc_b + 128, nullptr,
      nullptr, 0, P1, 128, NTOT, 128, 128, 0);
  gemm_f32<false, false, false, false><<<dim3(NTOT / 16, 1), 256, 0, stream>>>(
      P3, 128, enc_w + 2 * 16384, 128, nullptr, nullptr, enc_b + 256, nullptr,
      nullptr, 0, P2, 128, NTOT, 128, 128, 0);
  attn_kernel<<<dim3(25, BGRAPH), 32, 0, stream>>>(P0, P1, P2, P4);
  // u = attn@w3^T + b3 + XH
  gemm_f32<false, false, true, false><<<dim3(NTOT / 16, 1), 256, 0, stream>>>(
      P4, 128, enc_w + 3 * 16384, 128, nullptr, nullptr, enc_b + 384, nullptr,
      P3, 128, P5, 128, NTOT, 128, 128, 0);
  // xe = u + relu(u@enc_lw^T + enc_lb)
  gemm_f32<false, true, true, false><<<dim3(NTOT / 16, 1), 256, 0, stream>>>(
      P5, 128, enc_lw, 128, nullptr, nullptr, enc_lb, nullptr, P5, 128, P0, 128,
      NTOT, 128, 128, 0);
  // y = relu(xe@pma_lw^T + pma_lb)
  gemm_f32<false, true, false, false><<<dim3(NTOT / 16, 1), 256, 0, stream>>>(
      P0, 128, pma_lw, 128, nullptr, nullptr, pma_lb, nullptr, nullptr, 0, P1,
      128, NTOT, 128, 128, 0);
  // kp, vp
  gemm_f32<false, false, false, false><<<dim3(NTOT / 16, 1), 256, 0, stream>>>(
      P1, 128, pma_w + 1 * 16384, 128, nullptr, nullptr, pma_b + 128, nullptr,
      nullptr, 0, P2, 128, NTOT, 128, 128, 0);
  gemm_f32<false, false, false, false><<<dim3(NTOT / 16, 1), 256, 0, stream>>>(
      P1, 128, pma_w + 2 * 16384, 128, nullptr, nullptr, pma_b + 256, nullptr,
      nullptr, 0, P4, 128, NTOT, 128, 128, 0);
  pma_ctx_kernel<<<BGRAPH, 128, 0, stream>>>(P2, P4, pma_seed, pma_w, pma_b,
                                             ctx);
  // xp_pre = ctx@pma_w3^T + pma_b3 + seed (seed folded into epilogue bias2)
  gemm_f32<false, false, false, false><<<dim3(32, 1), 256, 0, stream>>>(
      ctx, 128, pma_w + 3 * 16384, 128, nullptr, nullptr, pma_b + 384, pma_seed,
      nullptr, 0, xp_pre, 128, 512, 128, 128, 0);
  // xp = xp_pre + relu(xp_pre@pma_mlw^T + pma_mlb)
  gemm_f32<false, true, true, false><<<dim3(32, 1), 256, 0, stream>>>(
      xp_pre, 128, pma_mlw, 128, nullptr, nullptr, pma_mlb, nullptr, xp_pre,
      128, xp, 128, 512, 128, 128, 0);
  // decoder SAB (1 token -> softmax==1 -> q/k dead): v, out+res, ff+res
  gemm_f32<false, false, false, false><<<dim3(32, 1), 256, 0, stream>>>(
      xp, 128, dec_w + 2 * 16384, 128, nullptr, nullptr, dec_b + 256, nullptr,
      nullptr, 0, vdec, 128, 512, 128, 128, 0);
  gemm_f32<false, false, true, false><<<dim3(32, 1), 256, 0, stream>>>(
      vdec, 128, dec_w + 3 * 16384, 128, nullptr, nullptr, dec_b + 384, nullptr,
      xp, 128, xs_pre, 128, 512, 128, 128, 0);
  gemm_f32<false, true, true, false><<<dim3(32, 1), 256, 0, stream>>>(
      xs_pre, 128, dec_lw, 128, nullptr, nullptr, dec_lb, nullptr, xs_pre, 128,
      pooled + 256, 384, 512, 128, 128, 0);

  // --- output head: out = pooled@out_w^T + out_b  (512 x 384, K=384) ---
  gemm_f32<false, false, false, false><<<dim3(32, 3), 256, 0, stream>>>(
      pooled, 384, out_w, 384, nullptr, nullptr, out_b, nullptr, nullptr, 0,
      (float*)d_out, 384, 512, 384, 384, 0);
}